// BottleneckedEncoder_12343736008845
// MI455X (gfx1250) — compile-verified
//
#include <hip/hip_runtime.h>
#include <hip/hip_bf16.h>

// ---------------------------------------------------------------------------
// Shapes (fixed by the reference)
// ---------------------------------------------------------------------------
#define CB    8        // codebooks
#define PP    2048     // pairs / codebook
#define DKV   256      // dim keys == dim values
#define BB    8192     // batch (bn = B*N = 8192)
#define TOPK  4
#define MROWS 65536    // CB * BB   (global row r = c*BB + t)

// d_out float offsets (qv, qk, keys_ind, dist, counts, flatten)
#define OFF_QV   0L
#define OFF_QK   67108864L
#define OFF_IND  134217728L
#define OFF_DIST 134479872L
#define OFF_CNT  134742016L
#define OFF_FLAT 135004160L

typedef __attribute__((ext_vector_type(2))) float    v2f;
typedef __attribute__((ext_vector_type(8))) float    v8f;
typedef __attribute__((ext_vector_type(4))) unsigned v4u;
typedef __attribute__((ext_vector_type(8))) int      v8i;
typedef __attribute__((ext_vector_type(4))) int      v4i;

__device__ __forceinline__ float warp_sum(float v) {
  #pragma unroll
  for (int off = 16; off > 0; off >>= 1) v += __shfl_xor(v, off, 32);
  return v;
}

// ---------------------------------------------------------------------------
// keys squared norms: kq[c*PP+p] = sum_k keys[c][p][k]^2
// ---------------------------------------------------------------------------
__global__ __launch_bounds__(256) void rowsq_kernel(const float* __restrict__ in,
                                                    float* __restrict__ out, int nrows) {
  const int wave = threadIdx.x >> 5, lane = threadIdx.x & 31;
  const int r = blockIdx.x * 8 + wave;
  if (r >= nrows) return;
  const float* src = in + (long)r * DKV;
  float s = 0.f;
  #pragma unroll
  for (int i = 0; i < 8; ++i) { float v = src[lane + 32 * i]; s += v * v; }
  s = warp_sum(s);
  if (lane == 0) out[r] = s;
}

// ---------------------------------------------------------------------------
// LayerNorm over dim 256, one wave per row.
// xmode=1: input is x (B,C,1,K): row r=(c,t) reads x[(t*CB+c)*K]  (folds the
//          'b c n k -> c (b n) k' transpose).  xmode=0: row-major input.
// ---------------------------------------------------------------------------
__global__ __launch_bounds__(256) void ln_kernel(const float* __restrict__ in,
                                                 const float* __restrict__ g,
                                                 const float* __restrict__ b,
                                                 float* __restrict__ out, int xmode) {
  const int wave = threadIdx.x >> 5, lane = threadIdx.x & 31;
  const long r = (long)blockIdx.x * 8 + wave;
  long inRow = r;
  if (xmode) { long c = r >> 13; long t = r & (BB - 1); inRow = t * CB + c; }
  const float* src = in + inRow * DKV;
  float vals[8]; float s = 0.f;
  #pragma unroll
  for (int i = 0; i < 8; ++i) { vals[i] = src[lane + 32 * i]; s += vals[i]; }
  s = warp_sum(s);
  const float mu = s * (1.f / 256.f);
  float sq = 0.f;
  #pragma unroll
  for (int i = 0; i < 8; ++i) { float d = vals[i] - mu; sq += d * d; }
  sq = warp_sum(sq);
  const float rs = rsqrtf(sq * (1.f / 256.f) + 1e-5f);
  float* dst = out + r * DKV;
  #pragma unroll
  for (int i = 0; i < 8; ++i) {
    int col = lane + 32 * i;
    dst[col] = (vals[i] - mu) * rs * g[col] + b[col];
  }
}

// ---------------------------------------------------------------------------
// Generic WMMA f32 GEMM:  Out[M x N] = A[M x 256] * W[N x 256]^T + bias
// Block tile 128x64, 8 waves, each wave 16 rows x 4 WMMA n-tiles.
// EPI: 0 = bias   1 = bias+GELU(exact)   2 = bias+Res[r*N+n]
//      3 = bias + Res gathered from x layout ((t*CB+c)*256+n)
// ---------------------------------------------------------------------------
template <int EPI>
__global__ __launch_bounds__(256) void gemm_kernel(const float* __restrict__ A,
                                                   const float* __restrict__ W,
                                                   const float* __restrict__ bias,
                                                   float* __restrict__ Out,
                                                   const float* __restrict__ Res,
                                                   int N) {
  __shared__ float As[128][36];   // stride 36 floats: conflict-free, 16B aligned rows
  __shared__ float Ws[64][36];

  const int tid = threadIdx.x;
  const int wave = tid >> 5, lane = tid & 31;
  const long m0 = (long)blockIdx.y * 128;
  const int  n0 = blockIdx.x * 64;

  v8f acc[4];
  #pragma unroll
  for (int j = 0; j < 4; ++j)
    #pragma unroll
    for (int i = 0; i < 8; ++i) acc[j][i] = 0.f;

  const int arow = tid >> 1, ahalf = tid & 1;   // A: 2 threads/row, 16 floats each
  const int wrow = tid >> 2, wq = tid & 3;      // W: 4 threads/row, 8 floats each
  const int rsel = lane & 15, ksel = (lane >> 4) * 2;

  for (int k0 = 0; k0 < 256; k0 += 32) {
    const float* ag = A + (m0 + arow) * 256 + k0 + ahalf * 16;
    #pragma unroll
    for (int i = 0; i < 4; ++i)
      *(float4*)&As[arow][ahalf * 16 + i * 4] = ((const float4*)ag)[i];
    const float* wg = W + (long)(n0 + wrow) * 256 + k0 + wq * 8;
    *(float4*)&Ws[wrow][wq * 8 + 0] = ((const float4*)wg)[0];
    *(float4*)&Ws[wrow][wq * 8 + 4] = ((const float4*)wg)[1];
    if (k0 + 32 < 256) {          // prefetch next K-step (global_prefetch_b8)
      __builtin_prefetch(ag + 32, 0, 0);
      __builtin_prefetch(wg + 32, 0, 0);
    }
    __syncthreads();

    #pragma unroll
    for (int kc = 0; kc < 32; kc += 4) {
      v2f a;
      a.x = As[16 * wave + rsel][kc + ksel];
      a.y = As[16 * wave + rsel][kc + ksel + 1];
      #pragma unroll
      for (int j = 0; j < 4; ++j) {
        v2f bf;
        bf.x = Ws[16 * j + rsel][kc + ksel];
        bf.y = Ws[16 * j + rsel][kc + ksel + 1];
        acc[j] = __builtin_amdgcn_wmma_f32_16x16x4_f32(false, a, false, bf,
                                                       (short)0, acc[j], false, false);
      }
    }
    __syncthreads();
  }

  const long rbase = m0 + wave * 16 + 8 * (lane >> 4);
  #pragma unroll
  for (int j = 0; j < 4; ++j) {
    const int n = n0 + j * 16 + (lane & 15);
    const float bv = bias[n];
    #pragma unroll
    for (int i = 0; i < 8; ++i) {
      const long r = rbase + i;
      float v = acc[j][i] + bv;
      if (EPI == 1) v = 0.5f * v * (1.f + erff(v * 0.70710678118654752f));
      else if (EPI == 2) v += Res[r * N + n];
      else if (EPI == 3) {
        long t = r & (BB - 1), c = r >> 13;
        v += Res[(t * CB + c) * 256 + n];
      }
      Out[r * N + n] = v;
    }
  }
}

// ---------------------------------------------------------------------------
// Attention across codebooks (L=8) per token/head; one wave per (token,head).
// ---------------------------------------------------------------------------
__global__ __launch_bounds__(256) void attn_kernel(const float* __restrict__ QKV,
                                                   float* __restrict__ O) {
  const int wave = threadIdx.x >> 5, lane = threadIdx.x & 31;
  const int item = blockIdx.x * 8 + wave;    // < BB*2
  const int t = item >> 1, h = item & 1;
  const int dbase = h * 128 + lane * 4;      // 128 dims / 32 lanes = 4 each
  float kr[8][4], vr[8][4];
  #pragma unroll
  for (int c = 0; c < 8; ++c) {
    const float* base = QKV + ((long)(c * BB + t)) * 768;
    float4 kk = *(const float4*)(base + 256 + dbase);
    float4 vv = *(const float4*)(base + 512 + dbase);
    kr[c][0] = kk.x; kr[c][1] = kk.y; kr[c][2] = kk.z; kr[c][3] = kk.w;
    vr[c][0] = vv.x; vr[c][1] = vv.y; vr[c][2] = vv.z; vr[c][3] = vv.w;
  }
  const float scale = 0.088388347648318447f;   // 1/sqrt(128)
  #pragma unroll
  for (int c = 0; c < 8; ++c) {
    const float* base = QKV + ((long)(c * BB + t)) * 768;
    float4 qq = *(const float4*)(base + dbase);
    float s[8];
    #pragma unroll
    for (int c2 = 0; c2 < 8; ++c2) {
      float p = qq.x * kr[c2][0] + qq.y * kr[c2][1] + qq.z * kr[c2][2] + qq.w * kr[c2][3];
      s[c2] = warp_sum(p) * scale;
    }
    float m = s[0];
    #pragma unroll
    for (int c2 = 1; c2 < 8; ++c2) m = fmaxf(m, s[c2]);
    float sum = 0.f;
    #pragma unroll
    for (int c2 = 0; c2 < 8; ++c2) { s[c2] = expf(s[c2] - m); sum += s[c2]; }
    const float inv = 1.f / sum;
    float4 o; o.x = o.y = o.z = o.w = 0.f;
    #pragma unroll
    for (int c2 = 0; c2 < 8; ++c2) {
      float a = s[c2] * inv;
      o.x += a * vr[c2][0]; o.y += a * vr[c2][1];
      o.z += a * vr[c2][2]; o.w += a * vr[c2][3];
    }
    *(float4*)(O + ((long)(c * BB + t)) * 256 + dbase) = o;
  }
}

// ---------------------------------------------------------------------------
// Fused  dist = 2*f.k - |f|^2 - |k|^2  (WMMA f32)  +  top-4  +  gathers.
// Block: (c, 64 tokens). Key tiles of 64 rows fetched by the Tensor Data
// Mover straight into padded LDS. Wave w: row-strip w&3, column-half w>>2
// (two n-tiles per A-fragment -> 1.5 LDS loads per WMMA).
// ---------------------------------------------------------------------------
__device__ __forceinline__ bool tk_better(float v, int i, float tv, int ti) {
  return (v > tv) || (v == tv && i < ti);
}
__device__ __forceinline__ void tk_insert(float v, int i,
    float& t0, int& i0, float& t1, int& i1,
    float& t2, int& i2, float& t3, int& i3) {
  if (tk_better(v, i, t3, i3)) {
    if (tk_better(v, i, t2, i2)) {
      t3 = t2; i3 = i2;
      if (tk_better(v, i, t1, i1)) {
        t2 = t1; i2 = i1;
        if (tk_better(v, i, t0, i0)) { t1 = t0; i1 = i0; t0 = v; i0 = i; }
        else { t1 = v; i1 = i; }
      } else { t2 = v; i2 = i; }
    } else { t3 = v; i3 = i; }
  }
}

__global__ __launch_bounds__(256) void dist_topk_kernel(
    const float* __restrict__ Fl,      // flatten, row (c*BB+t)*256
    const float* __restrict__ keys, const float* __restrict__ values,
    const float* __restrict__ counter, const float* __restrict__ kq,
    float* __restrict__ out) {
  __shared__ float Af[64][260];        // flatten tile
  __shared__ float Kt[64][260];        // key tile (TDM fills, 4-DWORD row pad)
  __shared__ float fsq[64];
  __shared__ float Dt[8][16][33];      // per-wave 16x32 dist scratch
  __shared__ float candV[8][16][4];
  __shared__ int   candI[8][16][4];

  const int tid = threadIdx.x, wave = tid >> 5, lane = tid & 31;
  const int c = blockIdx.y;
  const long t0 = (long)blockIdx.x * 64;

  { // load 64 flatten rows (4 threads/row, 64 floats each)
    const int row = tid >> 2, q = tid & 3;
    const float* src = Fl + ((long)c * BB + t0 + row) * 256 + q * 64;
    #pragma unroll
    for (int i = 0; i < 16; ++i)
      *(float4*)&Af[row][q * 64 + i * 4] = ((const float4*)src)[i];
  }
  __syncthreads();
  if (tid < 64) {
    float s = 0.f;
    for (int k = 0; k < 256; ++k) { float v = Af[tid][k]; s += v * v; }
    fsq[tid] = s;
  }

  float tv0 = -3e38f, tv1 = -3e38f, tv2 = -3e38f, tv3 = -3e38f;
  int   ti0 = 0x7fffffff, ti1 = 0x7fffffff, ti2 = 0x7fffffff, ti3 = 0x7fffffff;

  const int rstrip = wave & 3;           // 16-row strip
  const int chalf  = wave >> 2;          // 32-col half of the 64-key tile
  const int rsel = lane & 15, ksel = (lane >> 4) * 2;
  const int myrow = lane & 15, myhalf = lane >> 4;

  const unsigned      ldsKt   = (unsigned)(size_t)&Kt[0][0];
  const unsigned long keyBase = (unsigned long)(size_t)keys;

  for (int pt = 0; pt < 32; ++pt) {
    const int p0 = pt * 64;
    if (wave == 0) {
      // ---- Tensor Data Mover: 64 key rows (256 f32) -> Kt with row pad ----
      const unsigned long ga = keyBase + ((unsigned long)(c * PP + p0)) * 1024ul;
      v4u g0;
      g0[0] = 1u;                                   // count=1 (valid user D#)
      g0[1] = ldsKt;                                // lds_addr
      g0[2] = (unsigned)(ga & 0xffffffffu);         // global_addr[31:0]
      g0[3] = (unsigned)((ga >> 32) & 0x1ffffffu) | 0x80000000u;  // addr[56:32] | type=2
      v8i g1;
      g1[0] = (int)((2u << 16) | (1u << 20) | (7u << 22) | (3u << 25));
             // data_size=4B | pad_enable | pad_interval=256 DW | pad_amount=4 DW
      g1[1] = (int)(256u << 16);                    // tensor_dim0 = 256
      g1[2] = (int)(2048u << 16);                   // tensor_dim1 = 2048
      g1[3] = (int)(256u << 16);                    // tile_dim0 = 256
      g1[4] = 64;                                   // tile_dim1 = 64
      g1[5] = 256;                                  // tensor_dim0_stride = 256
      g1[6] = 0; g1[7] = 0;
      v4i g2; g2[0] = 0; g2[1] = 0; g2[2] = 0; g2[3] = 0;
      v4i g3 = g2;
#if __clang_major__ >= 23
      v8i g4; g4[0]=0; g4[1]=0; g4[2]=0; g4[3]=0; g4[4]=0; g4[5]=0; g4[6]=0; g4[7]=0;
      __builtin_amdgcn_tensor_load_to_lds(g0, g1, g2, g3, g4, 0);
#else
      __builtin_amdgcn_tensor_load_to_lds(g0, g1, g2, g3, 0);
#endif
      __builtin_amdgcn_s_wait_tensorcnt(0);
    }
    __syncthreads();

    v8f acc0, acc1;
    #pragma unroll
    for (int i = 0; i < 8; ++i) { acc0[i] = 0.f; acc1[i] = 0.f; }
    #pragma unroll 8
    for (int kc = 0; kc < 256; kc += 4) {
      v2f a, b0, b1;
      a.x  = Af[rstrip * 16 + rsel][kc + ksel];
      a.y  = Af[rstrip * 16 + rsel][kc + ksel + 1];
      b0.x = Kt[chalf * 32 + rsel][kc + ksel];
      b0.y = Kt[chalf * 32 + rsel][kc + ksel + 1];
      b1.x = Kt[chalf * 32 + 16 + rsel][kc + ksel];
      b1.y = Kt[chalf * 32 + 16 + rsel][kc + ksel + 1];
      acc0 = __builtin_amdgcn_wmma_f32_16x16x4_f32(false, a, false, b0,
                                                   (short)0, acc0, false, false);
      acc1 = __builtin_amdgcn_wmma_f32_16x16x4_f32(false, a, false, b1,
                                                   (short)0, acc1, false, false);
    }
    const float kq0 = kq[c * PP + p0 + chalf * 32 + rsel];
    const float kq1 = kq[c * PP + p0 + chalf * 32 + 16 + rsel];
    #pragma unroll
    for (int i = 0; i < 8; ++i) {
      const int rr = myhalf * 8 + i;
      const float fs = fsq[rstrip * 16 + rr];
      Dt[wave][rr][rsel]      = 2.f * acc0[i] - fs - kq0;
      Dt[wave][rr][rsel + 16] = 2.f * acc1[i] - fs - kq1;
    }
    __syncthreads();
    #pragma unroll
    for (int j2 = 0; j2 < 16; ++j2) {
      const int colc = myhalf * 16 + j2;           // 0..31 within wave's half
      tk_insert(Dt[wave][myrow][colc], p0 + chalf * 32 + colc,
                tv0, ti0, tv1, ti1, tv2, ti2, tv3, ti3);
    }
  }

  // intra-wave merge of the two lane-halves for each row
  float ov0 = __shfl(tv0, lane + 16, 32), ov1 = __shfl(tv1, lane + 16, 32);
  float ov2 = __shfl(tv2, lane + 16, 32), ov3 = __shfl(tv3, lane + 16, 32);
  int   oi0 = __shfl(ti0, lane + 16, 32), oi1 = __shfl(ti1, lane + 16, 32);
  int   oi2 = __shfl(ti2, lane + 16, 32), oi3 = __shfl(ti3, lane + 16, 32);
  if (lane < 16) {
    tk_insert(ov0, oi0, tv0, ti0, tv1, ti1, tv2, ti2, tv3, ti3);
    tk_insert(ov1, oi1, tv0, ti0, tv1, ti1, tv2, ti2, tv3, ti3);
    tk_insert(ov2, oi2, tv0, ti0, tv1, ti1, tv2, ti2, tv3, ti3);
    tk_insert(ov3, oi3, tv0, ti0, tv1, ti1, tv2, ti2, tv3, ti3);
    candV[wave][lane][0] = tv0; candI[wave][lane][0] = ti0;
    candV[wave][lane][1] = tv1; candI[wave][lane][1] = ti1;
    candV[wave][lane][2] = tv2; candI[wave][lane][2] = ti2;
    candV[wave][lane][3] = tv3; candI[wave][lane][3] = ti3;
  }
  __syncthreads();
  // cross-wave merge: wave w (<4) merges its regs with wave w+4's candidates
  if (wave < 4 && lane < 16) {
    #pragma unroll
    for (int j = 0; j < 4; ++j)
      tk_insert(candV[wave + 4][lane][j], candI[wave + 4][lane][j],
                tv0, ti0, tv1, ti1, tv2, ti2, tv3, ti3);
    candV[wave][lane][0] = tv0; candI[wave][lane][0] = ti0;
    candV[wave][lane][1] = tv1; candI[wave][lane][1] = ti1;
    candV[wave][lane][2] = tv2; candI[wave][lane][2] = ti2;
    candV[wave][lane][3] = tv3; candI[wave][lane][3] = ti3;
  }
  __syncthreads();

  // gather outputs: each wave emits 8 tokens (coalesced 1KB copies + scalars)
  for (int i = 0; i < 8; ++i) {
    const int tr = wave * 8 + i;                   // token 0..63 in block
    const int strip = tr >> 4, rowin = tr & 15;
    const long t = t0 + tr;
    #pragma unroll
    for (int j = 0; j < 4; ++j) {
      const int idx = candI[strip][rowin][j];
      const long ob = (((t * CB + c) * TOPK) + j) * 256;
      const float* ksrc = keys + ((long)c * PP + idx) * 256;
      const float* vsrc = values + ((long)c * PP + idx) * 256;
      *(float4*)(out + OFF_QK + ob + lane * 8)     = *(const float4*)(ksrc + lane * 8);
      *(float4*)(out + OFF_QK + ob + lane * 8 + 4) = *(const float4*)(ksrc + lane * 8 + 4);
      *(float4*)(out + OFF_QV + ob + lane * 8)     = *(const float4*)(vsrc + lane * 8);
      *(float4*)(out + OFF_QV + ob + lane * 8 + 4) = *(const float4*)(vsrc + lane * 8 + 4);
    }
    if (lane < 4) {
      const int j = lane;
      const int idx = candI[strip][rowin][j];
      const long sb = (t * CB + c) * TOPK + j;
      out[OFF_IND  + sb] = (float)idx;
      out[OFF_DIST + sb] = candV[strip][rowin][j];
      out[OFF_CNT  + sb] = counter[c * PP + idx];
    }
  }
}

// ---------------------------------------------------------------------------
extern "C" void kernel_launch(void* const* d_in, const int* in_sizes, int n_in,
                              void* d_out, int out_size, void* d_ws, size_t ws_size,
                              hipStream_t stream) {
  const float* x       = (const float*)d_in[0];
  const float* keys    = (const float*)d_in[1];
  const float* values  = (const float*)d_in[2];
  const float* counter = (const float*)d_in[3];
  const float* ln1_g   = (const float*)d_in[4];
  const float* ln1_b   = (const float*)d_in[5];
  const float* w_in    = (const float*)d_in[6];
  const float* b_in    = (const float*)d_in[7];
  const float* w_out   = (const float*)d_in[8];
  const float* b_out   = (const float*)d_in[9];
  const float* ln2_g   = (const float*)d_in[10];
  const float* ln2_b   = (const float*)d_in[11];
  const float* fw1     = (const float*)d_in[12];
  const float* fb1     = (const float*)d_in[13];
  const float* fw2     = (const float*)d_in[14];
  const float* fb2     = (const float*)d_in[15];
  const float* dw      = (const float*)d_in[16];
  const float* db      = (const float*)d_in[17];

  float* out = (float*)d_out;
  float* ws  = (float*)d_ws;

  float* L    = ws;                    // 16.78M  (ln1 out; reused as ln2 out)
  float* QKV  = ws + 16777216L;        // 50.33M  (reused as FFN mid)
  float* O    = ws + 67108864L;        // 16.78M  (attn out; reused as H2)
  float* H1   = ws + 83886080L;        // 16.78M
  float* KQ   = ws + 100663296L;       // 16K key sq-norms
  float* G    = L;
  float* F1   = QKV;
  float* H2   = O;
  float* FLAT = out + OFF_FLAT;

  rowsq_kernel<<<2048, 256, 0, stream>>>(keys, KQ, CB * PP);
  ln_kernel<<<MROWS / 8, 256, 0, stream>>>(x, ln1_g, ln1_b, L, 1);
  gemm_kernel<0><<<dim3(12, MROWS / 128), 256, 0, stream>>>(L, w_in, b_in, QKV, nullptr, 768);
  attn_kernel<<<(BB * 2) / 8, 256, 0, stream>>>(QKV, O);
  gemm_kernel<3><<<dim3(4, MROWS / 128), 256, 0, stream>>>(O, w_out, b_out, H1, x, 256);
  ln_kernel<<<MROWS / 8, 256, 0, stream>>>(H1, ln2_g, ln2_b, G, 0);
  gemm_kernel<1><<<dim3(4, MROWS / 128), 256, 0, stream>>>(G, fw1, fb1, F1, nullptr, 256);
  gemm_kernel<2><<<dim3(4, MROWS / 128), 256, 0, stream>>>(F1, fw2, fb2, H2, H1, 256);
  gemm_kernel<0><<<dim3(4, MROWS / 128), 256, 0, stream>>>(H2, dw, db, FLAT, nullptr, 256);
  dist_topk_kernel<<<dim3(BB / 64, CB), 256, 0, stream>>>(FLAT, keys, values, counter, KQ, out);
}